// KalmanFilter_50276887167553
// MI455X (gfx1250) — compile-verified
//
#include <hip/hip_runtime.h>
#include <hip/hip_bf16.h>

// Kalman filter, 32768 independent tracks, T=256 sequential steps.
// One thread per track. Observations streamed HBM -> LDS with the CDNA5
// async-load engine (ASYNCcnt), double buffered per 16-step chunk.

#define B_   128
#define T_   256
#define V_   256
#define NT   128                 // threads per block (one b, half of v)
#define TC   16                  // time steps per chunk
#define NC   (T_ / TC)           // 16 chunks
#define ROWF (V_ * 3)            // 768 floats per (b, t)
#define HALF (NT * 3)            // 384 floats per block per t
#define CH4  (TC * HALF / 4)     // 1536 float4 per chunk
#define LPT  (CH4 / NT)          // 12 async b128 loads per thread per chunk

typedef __attribute__((ext_vector_type(4))) int v4i;
typedef __attribute__((address_space(1))) v4i* g_v4i_ptr;
typedef __attribute__((address_space(3))) v4i* l_v4i_ptr;

// ---- CDNA5 async global->LDS load (ASYNCcnt-tracked) ------------------------

__device__ __forceinline__ void async_ld_b128(const float* src, unsigned lds_addr) {
#if __has_builtin(__builtin_amdgcn_global_load_async_to_lds_b128)
  __builtin_amdgcn_global_load_async_to_lds_b128(
      (g_v4i_ptr)(unsigned long long)src,
      (l_v4i_ptr)(unsigned long long)lds_addr,
      0, 0);
#else
  asm volatile("global_load_async_to_lds_b128 %0, %1, off"
               :: "v"(lds_addr), "v"(src) : "memory");
#endif
}

template <int N>
__device__ __forceinline__ void wait_asynccnt() {
#if __has_builtin(__builtin_amdgcn_s_wait_asynccnt)
  __builtin_amdgcn_s_wait_asynccnt(N);
#else
  asm volatile("s_wait_asynccnt %0" :: "n"(N));
#endif
}

// Issue all async loads for one chunk. 1536 float4, 12 per thread.
__device__ __forceinline__ void issue_chunk(const float* gptr, int chunk,
                                            float (*dst)[HALF], int tid) {
  const float* cbase = gptr + (size_t)chunk * TC * ROWF;
#pragma unroll
  for (int i = 0; i < LPT; ++i) {
    int jj = tid + NT * i;       // 0..1535
    int t  = jj / 96;            // 96 float4 per time step (const divisor)
    int r  = jj - t * 96;
    const float* src = cbase + (size_t)t * ROWF + r * 4;
    unsigned lds = (unsigned)(unsigned long long)(&dst[t][r * 4]);
    async_ld_b128(src, lds);
  }
}

// ---- One Kalman step (symmetric P: a..j = P00,P01,P02,P03,P11,P12,P13,P22,P23,P33)
// Masking trick: where(mask, upd, pred) == update with K zeroed when masked,
// since s_upd = s_pred + K*y and P_upd = P_pred - K*(rows01 of P_pred).
__device__ __forceinline__ void kstep(float lab, float z0, float z1,
    float& x, float& y, float& vx, float& vy,
    float& a, float& b, float& c, float& d, float& e,
    float& f, float& g, float& h, float& i2, float& j) {
  const float q = 0.01f;
  // predict: s = F s ; P = F P F^T + Q  (F: x+=vx, y+=vy)
  float xp = x + vx, yp = y + vy;
  float A  = a + 2.0f * c + h + q;   // P00
  float Bq = b + d + f + i2;         // P01
  float C  = c + h;                  // P02
  float D  = d + i2;                 // P03
  float E  = e + 2.0f * g + j + q;   // P11
  float Fv = f + i2;                 // P12
  float G  = g + j;                  // P13
  float H2 = h + q;                  // P22
  float I3 = i2;                     // P23
  float J  = j + q;                  // P33
  // innovation covariance S = P[:2,:2] + I, closed-form 2x2 inverse
  float s00 = A + 1.0f, s01 = Bq, s11 = E + 1.0f;
  float det  = s00 * s11 - s01 * s01;
  float rdet = __builtin_amdgcn_rcpf(det);
  const bool m = (lab != -1.0f);
  rdet = m ? rdet : 0.0f;            // masked step => K == 0 (8 gains via 1+8 muls)
  // K = P[:, :2] inv(S)
  float k00 = (A  * s11 - Bq * s01) * rdet;
  float k01 = (Bq * s00 - A  * s01) * rdet;
  float k10 = (Bq * s11 - E  * s01) * rdet;
  float k11 = (E  * s00 - Bq * s01) * rdet;
  float k20 = (C  * s11 - Fv * s01) * rdet;
  float k21 = (Fv * s00 - C  * s01) * rdet;
  float k30 = (D  * s11 - G  * s01) * rdet;
  float k31 = (G  * s00 - D  * s01) * rdet;
  // state update (y zeroed by K==0 when masked)
  float y0 = z0 - xp, y1 = z1 - yp;
  x  = xp + k00 * y0 + k01 * y1;
  y  = yp + k10 * y0 + k11 * y1;
  vx = vx + k20 * y0 + k21 * y1;
  vy = vy + k30 * y0 + k31 * y1;
  // P = (I - K H) P_pred  (rows 0,1 of P_pred enter via H)
  a  = A  - (k00 * A  + k01 * Bq);
  b  = Bq - (k00 * Bq + k01 * E);
  c  = C  - (k00 * C  + k01 * Fv);
  d  = D  - (k00 * D  + k01 * G);
  e  = E  - (k10 * Bq + k11 * E);
  f  = Fv - (k10 * C  + k11 * Fv);
  g  = G  - (k10 * D  + k11 * G);
  h  = H2 - (k20 * C  + k21 * Fv);
  i2 = I3 - (k20 * D  + k21 * G);
  j  = J  - (k30 * D  + k31 * G);
}

__global__ __launch_bounds__(NT)
void KalmanFilter_50276887167553_kernel(const float* __restrict__ batch,
                                        float* __restrict__ out) {
  __shared__ alignas(16) float buf[2][TC][HALF];   // 48 KB double buffer
  const int tid = threadIdx.x;
  const int blk = blockIdx.x;          // 256 blocks: (b, v-half)
  const int b   = blk >> 1;
  const int v0  = (blk & 1) * NT;
  const float* gptr = batch + ((size_t)b * T_) * ROWF + (size_t)v0 * 3;

  // initial state: s = 0, P = 1000 I
  float x = 0.f, y = 0.f, vx = 0.f, vy = 0.f;
  float a = 1000.f, bb = 0.f, c = 0.f, d = 0.f, e = 1000.f,
        f = 0.f, g = 0.f, h = 1000.f, i2 = 0.f, j = 1000.f;

  issue_chunk(gptr, 0, buf[0], tid);
  for (int k = 0; k < NC; ++k) {
    const int cur = k & 1;
    if (k + 1 < NC) {
      issue_chunk(gptr, k + 1, buf[cur ^ 1], tid);   // overlap next chunk
      wait_asynccnt<LPT>();   // chunk k retired (async loads in-order per wave)
    } else {
      wait_asynccnt<0>();
    }
    __syncthreads();          // chunk k visible to whole block
#pragma unroll 4
    for (int lt = 0; lt < TC; ++lt) {
      const float* p = &buf[cur][lt][tid * 3];
      kstep(p[0], p[1], p[2], x, y, vx, vy, a, bb, c, d, e, f, g, h, i2, j);
    }
    __syncthreads();          // buf[cur] free for chunk k+2's async writes
  }

  float* o = out + ((size_t)b * V_ + v0 + tid) * 3;
  o[0] = 1.0f; o[1] = x; o[2] = y;
}

extern "C" void kernel_launch(void* const* d_in, const int* in_sizes, int n_in,
                              void* d_out, int out_size, void* d_ws, size_t ws_size,
                              hipStream_t stream) {
  const float* batch = (const float*)d_in[0];
  float* out = (float*)d_out;
  hipLaunchKernelGGL(KalmanFilter_50276887167553_kernel,
                     dim3(B_ * 2), dim3(NT), 0, stream, batch, out);
}